// MultiResHashEncoder_34239479283957
// MI455X (gfx1250) — compile-verified
//
#include <hip/hip_runtime.h>
#include <stdint.h>

// Multi-resolution hash-grid encoder (Instant-NGP style) for MI455X (gfx1250).
//
// Perf model: ~1 GFLOP total vs 96M random 8B L2 gathers + 96MB streamed
// output -> gather-request bound. fp32 scalar path (WMMA inapplicable: the
// blend is a per-lane block-diagonal 1x8 * 8x2 contraction, not a wave-shared
// matrix). Data movement tuned for CDNA5:
//   - coordinate tile (3KB/block) staged into LDS by the Tensor Data Mover
//     (TENSOR_LOAD_TO_LDS, 1D descriptor, OOB dwords zero-filled by TDM)
//   - 96 b64 gathers/point from L2-resident tables (12.6MB << 192MB L2)
//   - non-temporal b128 output stores (write-once 96MB stream)

typedef __attribute__((ext_vector_type(2))) float v2f;
typedef __attribute__((ext_vector_type(4))) float v4f;
typedef __attribute__((ext_vector_type(4))) unsigned int v4u;
typedef __attribute__((ext_vector_type(4))) int v4i;
typedef __attribute__((ext_vector_type(8))) int v8i;

typedef __attribute__((address_space(3))) const float* lds_cfp;

#define NLEV 12
#define HASH_SIZE (1u << 17)
#define HASH_MASK (HASH_SIZE - 1u)
#define BLOCK 256

__global__ __launch_bounds__(BLOCK) void MultiResHashEncoder_kernel(
    const float* __restrict__ x,       // [N,3]
    const float* __restrict__ tables,  // [L, HASH_SIZE, 2]
    float* __restrict__ out,           // [N, L, 2]
    int n)
{
  __shared__ float xt[3 * BLOCK];  // 3KB coordinate tile; only LDS object -> offset 0

  const int tid = threadIdx.x;
  const int i = blockIdx.x * BLOCK + tid;

  // ---- TDM: DMA this block's 256x3 f32 coordinate tile into LDS ----------
  if ((tid >> 5) == 0) {  // wave 0 only; TDM is a per-wave scalar DMA op
    const uint64_t gaddr =
        (uint64_t)(const void*)(x + (size_t)blockIdx.x * (3 * BLOCK));
    const uint32_t td0 = 3u * (uint32_t)n;  // tensor_dim0 (dwords); OOB reads -> 0
    const uint32_t tile0 = 3u * BLOCK;      // 768 dwords per block tile

    // D# group 0: count=1 | lds_addr | global_addr[56:0] | type=2
    v4u g0;
    g0.x = 1u;
    g0.y = 0u;                                 // lds_addr (bytes) = 0
    g0.z = (uint32_t)(gaddr & 0xFFFFFFFFull);  // global_addr[31:0]
    g0.w = (uint32_t)((gaddr >> 32) & 0x1FFFFFFull) | (2u << 30);  // [56:32]|type

    // D# group 1: data_size=4B, tensor_dim0=3N, tile_dim0=768, 1-D tile
    v8i g1;
    g1[0] = (int)(2u << 16);                               // data_size code 2 = 4B
    g1[1] = (int)((td0 & 0xFFFFu) << 16);                  // tensor_dim0[15:0]
    g1[2] = (int)(((td0 >> 16) & 0xFFFFu) | (1u << 16));   // dim0[31:16] | tensor_dim1=1
    g1[3] = (int)(tile0 << 16);                            // tile_dim0 = 768
    g1[4] = 0;                                             // tile_dim1=0, tile_dim2=0
    g1[5] = (int)td0;                                      // dim0_stride (unused, 1D)
    g1[6] = 0;
    g1[7] = 0;

    v4i g2 = {0, 0, 0, 0};
    v4i g3 = {0, 0, 0, 0};
    v8i g4 = {0, 0, 0, 0, 0, 0, 0, 0};  // extra group, 6-arg clang-23 form

    __builtin_amdgcn_tensor_load_to_lds(g0, g1, g2, g3, g4, 0);
    __builtin_amdgcn_s_wait_tensorcnt(0);
  }
  __syncthreads();

  // The TDM builtin's LDS write isn't modeled as a store to `xt`, so launder
  // an addrspace(3) pointer through an asm with a memory clobber: escapes the
  // LDS address (keeps the loads alive, no poison-DCE) while staying in the
  // LDS address space so the reads lower to native ds_load (not flat).
  lds_cfp xs = (lds_cfp)&xt[0];
  asm volatile("" : "+v"(xs)::"memory");

  // Stride-3-dword per-lane LDS reads: conflict-free on 64 banks.
  const float px = xs[3 * tid + 0];
  const float py = xs[3 * tid + 1];
  const float pz = xs[3 * tid + 2];

  // max(1, round(16 * b^l)), b = exp((ln256-ln16)/11)  (matches reference).
  const float res[NLEV] = {16.f, 21.f, 26.f, 34.f, 44.f, 56.f,
                           73.f, 93.f, 120.f, 155.f, 199.f, 256.f};

  float acc[2 * NLEV];

#pragma unroll
  for (int l = 0; l < NLEV; ++l) {
    const float r = res[l];
    const float sx = px * r, sy = py * r, sz = pz * r;
    const float fx = floorf(sx), fy = floorf(sy), fz = floorf(sz);
    const float wx = sx - fx, wy = sy - fy, wz = sz - fz;
    const int ix = (int)fx, iy = (int)fy, iz = (int)fz;

    // Spatial hash, int32-wraparound semantics; (c+1)*P = c*P + P.
    const uint32_t hx0 = (uint32_t)ix * 73856093u;
    const uint32_t hx1 = hx0 + 73856093u;
    const uint32_t hy0 = (uint32_t)iy * 19349663u;
    const uint32_t hy1 = hy0 + 19349663u;
    const uint32_t hz0 = (uint32_t)iz * 83492791u;
    const uint32_t hz1 = hz0 + 83492791u;

    const v2f* __restrict__ tbl = (const v2f*)tables + (size_t)l * HASH_SIZE;

    // 8 independent b64 gathers; indices always masked -> safe for all lanes.
    const v2f f000 = tbl[(hx0 ^ hy0 ^ hz0) & HASH_MASK];
    const v2f f001 = tbl[(hx0 ^ hy0 ^ hz1) & HASH_MASK];
    const v2f f010 = tbl[(hx0 ^ hy1 ^ hz0) & HASH_MASK];
    const v2f f011 = tbl[(hx0 ^ hy1 ^ hz1) & HASH_MASK];
    const v2f f100 = tbl[(hx1 ^ hy0 ^ hz0) & HASH_MASK];
    const v2f f101 = tbl[(hx1 ^ hy0 ^ hz1) & HASH_MASK];
    const v2f f110 = tbl[(hx1 ^ hy1 ^ hz0) & HASH_MASK];
    const v2f f111 = tbl[(hx1 ^ hy1 ^ hz1) & HASH_MASK];

    // Factored trilinear weights: 12 muls instead of 16.
    const float ux0 = 1.f - wx, uy0 = 1.f - wy, uz0 = 1.f - wz;
    const float w00 = ux0 * uy0, w01 = ux0 * wy;
    const float w10 = wx * uy0, w11 = wx * wy;
    const float w000 = w00 * uz0, w001 = w00 * wz;
    const float w010 = w01 * uz0, w011 = w01 * wz;
    const float w100 = w10 * uz0, w101 = w10 * wz;
    const float w110 = w11 * uz0, w111 = w11 * wz;

    float a0 = w000 * f000.x;
    float a1 = w000 * f000.y;
    a0 = fmaf(w001, f001.x, a0); a1 = fmaf(w001, f001.y, a1);
    a0 = fmaf(w010, f010.x, a0); a1 = fmaf(w010, f010.y, a1);
    a0 = fmaf(w011, f011.x, a0); a1 = fmaf(w011, f011.y, a1);
    a0 = fmaf(w100, f100.x, a0); a1 = fmaf(w100, f100.y, a1);
    a0 = fmaf(w101, f101.x, a0); a1 = fmaf(w101, f101.y, a1);
    a0 = fmaf(w110, f110.x, a0); a1 = fmaf(w110, f110.y, a1);
    a0 = fmaf(w111, f111.x, a0); a1 = fmaf(w111, f111.y, a1);

    acc[2 * l + 0] = a0;
    acc[2 * l + 1] = a1;
  }

  // 24 contiguous floats per point -> 6 non-temporal b128 stores.
  if (i < n) {
    v4f* o = (v4f*)(out + (size_t)i * (2 * NLEV));
#pragma unroll
    for (int q = 0; q < (2 * NLEV) / 4; ++q) {
      v4f v = {acc[4 * q + 0], acc[4 * q + 1], acc[4 * q + 2], acc[4 * q + 3]};
      __builtin_nontemporal_store(v, o + q);
    }
  }
}

extern "C" void kernel_launch(void* const* d_in, const int* in_sizes, int n_in,
                              void* d_out, int out_size, void* d_ws, size_t ws_size,
                              hipStream_t stream) {
  const float* x = (const float*)d_in[0];       // [N,3] f32
  const float* tables = (const float*)d_in[1];  // [12, 2^17, 2] f32
  float* out = (float*)d_out;                   // [N, 12, 2] f32

  const int n = in_sizes[0] / 3;
  const int blocks = (n + 255) / 256;
  hipLaunchKernelGGL(MultiResHashEncoder_kernel, dim3(blocks), dim3(256), 0,
                     stream, x, tables, out, n);
}